// GSAGE_15590731285056
// MI455X (gfx1250) — compile-verified
//
#include <hip/hip_runtime.h>

typedef float v2f __attribute__((ext_vector_type(2)));
typedef float v8f __attribute__((ext_vector_type(8)));

#define BM 64
#define BN 64
#define BK 16
#define ASTR (BK + 2)   // 18: keeps (k,k+1) pairs 8B-aligned; 18*r mod 64 distinct over 16 rows
#define BSTR 160        // floats per B pair-row; 160 % 64 == 32 -> the two half-waves use disjoint banks

// ---------------------------------------------------------------- utilities
__global__ void zero_f32(float* __restrict__ p, long n) {
    long i = (long)blockIdx.x * blockDim.x + threadIdx.x;
    if (i < n) p[i] = 0.0f;
}

__global__ void deg_kernel(const int* __restrict__ edge_dst, float* __restrict__ deg, int E) {
    int i = blockIdx.x * blockDim.x + threadIdx.x;
    if (i < E) atomicAdd(&deg[edge_dst[i]], 1.0f);
}

__global__ void invdeg_kernel(float* __restrict__ deg, int n) {
    int i = blockIdx.x * blockDim.x + threadIdx.x;
    if (i < n) deg[i] = 1.0f / fmaxf(deg[i], 1.0f);
}

// scatter-add h[src] rows into agg[dst]; one thread per (edge, feature)
__global__ void scatter_edges(const float* __restrict__ h, const int* __restrict__ src,
                              const int* __restrict__ dst, float* __restrict__ agg,
                              int E, int F) {
    long t = (long)blockIdx.x * blockDim.x + threadIdx.x;
    long total = (long)E * F;
    if (t >= total) return;
    int e = (int)(t / F);
    int f = (int)(t - (long)e * F);
    float v = h[(long)src[e] * F + f];
    atomicAdd(&agg[(long)dst[e] * F + f], v);
}

// vectorized variant: one thread per (edge, 4 features); F must be a multiple of 4
__global__ void scatter_edges_vec4(const float* __restrict__ h, const int* __restrict__ src,
                                   const int* __restrict__ dst, float* __restrict__ agg,
                                   int E, int F) {
    long t = (long)blockIdx.x * blockDim.x + threadIdx.x;
    const int Fq = F >> 2;
    long total = (long)E * Fq;
    if (t >= total) return;
    int e  = (int)(t / Fq);
    int f4 = (int)(t - (long)e * Fq) << 2;
    const float4 v = *(const float4*)&h[(long)src[e] * F + f4];
    float* a = &agg[(long)dst[e] * F + f4];
    atomicAdd(a + 0, v.x);
    atomicAdd(a + 1, v.y);
    atomicAdd(a + 2, v.z);
    atomicAdd(a + 3, v.w);
}

// ------------------------------------------------- fused SAGE layer (WMMA)
// Out[M,Nout] = act( H @ Wself + (AGG * invdeg) @ Wneigh + bias )
template <int RELU>
__global__ __launch_bounds__(128)
void sage_layer_gemm(const float* __restrict__ H, const float* __restrict__ AGG,
                     const float* __restrict__ invdeg,
                     const float* __restrict__ Wself, const float* __restrict__ Wneigh,
                     const float* __restrict__ bias, float* __restrict__ Out,
                     int M, int K, int Nout) {
    __shared__ float sH[BM * ASTR];
    __shared__ float sA[BM * ASTR];
    __shared__ float sBs[(BK / 2) * BSTR];   // (k,n) -> [(k/2)*BSTR + 2n + (k&1)]
    __shared__ float sBn[(BK / 2) * BSTR];

    const int tid  = threadIdx.x;
    const int wave = tid >> 5;
    const int lane = tid & 31;
    const int half = lane >> 4;   // 0: lanes 0-15 (K=kk,kk+1), 1: lanes 16-31 (K=kk+2,kk+3)
    const int l16  = lane & 15;
    const int m0   = blockIdx.x * BM;
    const int n0   = blockIdx.y * BN;

    v8f acc[4];
    #pragma unroll
    for (int j = 0; j < 4; ++j) {
        float b = bias[n0 + j * 16 + l16];
        #pragma unroll
        for (int v = 0; v < 8; ++v) acc[j][v] = b;
    }

    const bool vecA = ((K & 3) == 0);
    const int kChunks = (K + BK - 1) / BK;
    for (int kc = 0; kc < kChunks; ++kc) {
        const int k0 = kc * BK;

        // ---- stage A tiles: BM x BK (h, agg*invdeg)
        if (vecA && (k0 + BK) <= K) {
            for (int i = tid; i < BM * BK / 4; i += 128) {
                int r = i >> 2, c4 = (i & 3) * 4;
                int gm = m0 + r;
                float4 hv = make_float4(0.f, 0.f, 0.f, 0.f);
                float4 av = hv;
                if (gm < M) {
                    hv = *(const float4*)&H[(long)gm * K + k0 + c4];
                    float4 t = *(const float4*)&AGG[(long)gm * K + k0 + c4];
                    float id = invdeg[gm];
                    av = make_float4(t.x * id, t.y * id, t.z * id, t.w * id);
                }
                float* ph = &sH[r * ASTR + c4];
                float* pa = &sA[r * ASTR + c4];
                ph[0] = hv.x; ph[1] = hv.y; ph[2] = hv.z; ph[3] = hv.w;
                pa[0] = av.x; pa[1] = av.y; pa[2] = av.z; pa[3] = av.w;
            }
        } else {
            for (int i = tid; i < BM * BK; i += 128) {
                int r = i >> 4, c = i & 15;
                int gm = m0 + r, gk = k0 + c;
                float hv = 0.f, av = 0.f;
                if (gm < M && gk < K) {
                    hv = H[(long)gm * K + gk];
                    av = AGG[(long)gm * K + gk] * invdeg[gm];
                }
                sH[r * ASTR + c] = hv;
                sA[r * ASTR + c] = av;
            }
        }

        // ---- stage B tiles pair-interleaved (always float4-safe: Nout % 64 == 0)
        for (int i = tid; i < BK * BN / 4; i += 128) {
            int r = i >> 4, c4 = (i & 15) * 4;
            int gk = k0 + r;
            float4 ws = make_float4(0.f, 0.f, 0.f, 0.f), wn = ws;
            if (gk < K) {
                ws = *(const float4*)&Wself[(long)gk * Nout + n0 + c4];
                wn = *(const float4*)&Wneigh[(long)gk * Nout + n0 + c4];
            }
            int base = (r >> 1) * BSTR + 2 * c4 + (r & 1);
            sBs[base + 0] = ws.x; sBs[base + 2] = ws.y; sBs[base + 4] = ws.z; sBs[base + 6] = ws.w;
            sBn[base + 0] = wn.x; sBn[base + 2] = wn.y; sBn[base + 4] = wn.z; sBn[base + 6] = wn.w;
        }
        __syncthreads();

        const int arow = wave * 16 + l16;
        #pragma unroll
        for (int kk = 0; kk < BK; kk += 4) {
            const int koff = kk + 2 * half;
            const int p    = koff >> 1;          // pair row: rows (koff, koff+1)
            v2f ah = *(const v2f*)&sH[arow * ASTR + koff];
            v2f aa = *(const v2f*)&sA[arow * ASTR + koff];
            v2f bsv[4], bnv[4];
            #pragma unroll
            for (int j = 0; j < 4; ++j) {
                bsv[j] = *(const v2f*)&sBs[p * BSTR + 2 * (j * 16 + l16)];
                bnv[j] = *(const v2f*)&sBn[p * BSTR + 2 * (j * 16 + l16)];
            }
            #pragma unroll
            for (int j = 0; j < 4; ++j)
                acc[j] = __builtin_amdgcn_wmma_f32_16x16x4_f32(
                    false, ah, false, bsv[j], (short)0, acc[j], false, false);
            #pragma unroll
            for (int j = 0; j < 4; ++j)
                acc[j] = __builtin_amdgcn_wmma_f32_16x16x4_f32(
                    false, aa, false, bnv[j], (short)0, acc[j], false, false);
        }
        __syncthreads();
    }

    // epilogue: C/D layout -> row = wave*16 + half*8 + v, col = n0 + j*16 + l16
    #pragma unroll
    for (int j = 0; j < 4; ++j) {
        #pragma unroll
        for (int v = 0; v < 8; ++v) {
            int row = m0 + wave * 16 + half * 8 + v;
            int col = n0 + j * 16 + l16;
            float val = acc[j][v];
            if (RELU) val = fmaxf(val, 0.0f);
            if (row < M) Out[(long)row * Nout + col] = val;
        }
    }
}

// ---------------------------------------------------------------- pooling
__global__ void pool_count(const int* __restrict__ gid, float* __restrict__ counts, int n) {
    int i = blockIdx.x * blockDim.x + threadIdx.x;
    if (i < n) atomicAdd(&counts[gid[i]], 1.0f);
}

__global__ void pool_accum(const float* __restrict__ h, const int* __restrict__ gid,
                           float* __restrict__ hg, int n, int F) {
    long t = (long)blockIdx.x * blockDim.x + threadIdx.x;
    long total = (long)n * F;
    if (t >= total) return;
    int node = (int)(t / F);
    int f    = (int)(t - (long)node * F);
    atomicAdd(&hg[(long)gid[node] * F + f], h[t]);
}

__global__ void pool_final(float* __restrict__ hg, const float* __restrict__ counts,
                           int G, int F) {
    int t = blockIdx.x * blockDim.x + threadIdx.x;
    if (t < G * F) hg[t] /= fmaxf(counts[t / F], 1.0f);
}

// ---------------------------------------------------------------- launcher
extern "C" void kernel_launch(void* const* d_in, const int* in_sizes, int n_in,
                              void* d_out, int out_size, void* d_ws, size_t ws_size,
                              hipStream_t stream) {
    const float* feature = (const float*)d_in[0];
    const int*   esrc    = (const int*)d_in[1];
    const int*   edst    = (const int*)d_in[2];
    const int*   gid     = (const int*)d_in[3];
    const float* Ws0 = (const float*)d_in[4];
    const float* Wn0 = (const float*)d_in[5];
    const float* b0  = (const float*)d_in[6];
    const float* Ws1 = (const float*)d_in[7];
    const float* Wn1 = (const float*)d_in[8];
    const float* b1  = (const float*)d_in[9];
    const float* Ws2 = (const float*)d_in[10];
    const float* Wn2 = (const float*)d_in[11];
    const float* b2  = (const float*)d_in[12];

    const int E   = in_sizes[1];
    const int N   = in_sizes[3];
    const int IN  = in_sizes[0] / N;   // 27
    const int HID = in_sizes[6];       // 512
    const int OUT = in_sizes[12];      // 256
    const int G   = 32;

    float* ws     = (float*)d_ws;
    float* invdeg = ws;                      // N floats (deg then inverted in place)
    float* agg    = ws + 20224;              // N*HID floats
    float* h1     = agg + (long)N * HID;     // N*HID floats
    float* h2     = h1 + (long)N * HID;      // N*HID floats
    float* counts = agg;                     // G floats, reused after layer-2 GEMM

    float* out_h  = (float*)d_out;           // N*OUT
    float* out_hg = out_h + (long)N * OUT;   // G*OUT

    const int T = 256;
    auto blocks = [](long n, int t) { return (int)((n + t - 1) / t); };
    auto scatter = [&](const float* h, int F) {
        if ((F & 3) == 0)
            scatter_edges_vec4<<<blocks((long)E * (F >> 2), T), T, 0, stream>>>(h, esrc, edst, agg, E, F);
        else
            scatter_edges<<<blocks((long)E * F, T), T, 0, stream>>>(h, esrc, edst, agg, E, F);
    };

    // degrees -> inv_deg
    zero_f32<<<blocks(N, T), T, 0, stream>>>(invdeg, N);
    deg_kernel<<<blocks(E, T), T, 0, stream>>>(edst, invdeg, E);
    invdeg_kernel<<<blocks(N, T), T, 0, stream>>>(invdeg, N);

    // ---- layer 0: [N,27] -> [N,512], ReLU
    zero_f32<<<blocks((long)N * IN, T), T, 0, stream>>>(agg, (long)N * IN);
    scatter(feature, IN);
    {
        dim3 grid((N + BM - 1) / BM, HID / BN);
        sage_layer_gemm<1><<<grid, 128, 0, stream>>>(feature, agg, invdeg, Ws0, Wn0, b0, h1, N, IN, HID);
    }

    // ---- layer 1: [N,512] -> [N,512], ReLU
    zero_f32<<<blocks((long)N * HID, T), T, 0, stream>>>(agg, (long)N * HID);
    scatter(h1, HID);
    {
        dim3 grid((N + BM - 1) / BM, HID / BN);
        sage_layer_gemm<1><<<grid, 128, 0, stream>>>(h1, agg, invdeg, Ws1, Wn1, b1, h2, N, HID, HID);
    }

    // ---- layer 2: [N,512] -> [N,256], no activation, write to d_out
    zero_f32<<<blocks((long)N * HID, T), T, 0, stream>>>(agg, (long)N * HID);
    scatter(h2, HID);
    {
        dim3 grid((N + BM - 1) / BM, OUT / BN);
        sage_layer_gemm<0><<<grid, 128, 0, stream>>>(h2, agg, invdeg, Ws2, Wn2, b2, out_h, N, HID, OUT);
    }

    // ---- per-graph mean pooling
    zero_f32<<<blocks((long)G * OUT, T), T, 0, stream>>>(out_hg, (long)G * OUT);
    zero_f32<<<blocks(G, T), T, 0, stream>>>(counts, G);
    pool_count<<<blocks(N, T), T, 0, stream>>>(gid, counts, N);
    pool_accum<<<blocks((long)N * OUT, T), T, 0, stream>>>(out_h, gid, out_hg, N, OUT);
    pool_final<<<blocks((long)G * OUT, T), T, 0, stream>>>(out_hg, counts, G, OUT);
}